// EncodeMolMPN_88553635709673
// MI455X (gfx1250) — compile-verified
//
#include <hip/hip_runtime.h>
#include <hip/hip_bf16.h>

// MPN encoder for MI455X (gfx1250), wave32, f32 WMMA 16x16x4.
//
// ws layout (floats): [ h : E*128 ][ s : N*128 ][ nodeW1 : N*128 ]
//   total = 600000*128 + 2*100000*128 = 102.4M floats = 409.6 MB
//
// All GEMMs use V_WMMA_F32_16X16X4_F32. A fragment (16x4 f32): lane%16 = M,
// VGPR pair holds K = {0,1} (lanes 0-15) or {2,3} (lanes 16-31). B fragment
// (4x16): lane%16 = N, VGPR pair holds K = {0,1}/{2,3}. D (16x16): vgpr v,
// row = v + 8*(lane/16), col = lane%16.

typedef __attribute__((ext_vector_type(2))) float v2f;
typedef __attribute__((ext_vector_type(8))) float v8f;

__device__ __forceinline__ v8f wmma4(v2f a, v2f b, v8f c) {
  return __builtin_amdgcn_wmma_f32_16x16x4_f32(
      /*neg_a=*/false, a, /*neg_b=*/false, b,
      /*c_mod=*/(short)0, c, /*reuse_a=*/false, /*reuse_b=*/false);
}

#define EH 128
#define TM 16
// padded LDS strides (bank = f(stride%64) -> conflict-free fragment reads)
#define S132 132   // 132%64==4 : (4*row + k) % 64 all distinct over 16 rows
#define S68  68    // 68%64 ==4 : same property, K=64 matrices
#define S20  20    // 20%64==20 : 20n mod 64 distinct over n=0..15 (gcd=4)
#define S130 130   // epilogue nodeW1 tile

// ---------------------------------------------------------------- utilities
__global__ void zero_kernel(float* __restrict__ p, long n) {
  long i = (long)blockIdx.x * blockDim.x + threadIdx.x;
  long stride = (long)gridDim.x * blockDim.x;
  for (; i < n; i += stride) p[i] = 0.0f;
}

__global__ void relu_kernel(const float* __restrict__ a, float* __restrict__ o, long n) {
  long i = (long)blockIdx.x * blockDim.x + threadIdx.x;
  long stride = (long)gridDim.x * blockDim.x;
  for (; i < n; i += stride) {
    float v = a[i];
    o[i] = v > 0.0f ? v : 0.0f;
  }
}

// s[from[e]][c] += h[e][c]   (flat over E*128)
__global__ void scatter_kernel(const float* __restrict__ h, const int* __restrict__ from,
                               float* __restrict__ s, long total) {
  long i = (long)blockIdx.x * blockDim.x + threadIdx.x;
  long stride = (long)gridDim.x * blockDim.x;
  for (; i < total; i += stride) {
    long e = i >> 7;
    int c = (int)(i & 127);
    atomicAdd(&s[(long)from[e] * EH + c], h[i]);
  }
}

// ------------------------------------------- node GEMM: out[M,128] = A[M,64]@B[64,128]+bias
__global__ __launch_bounds__(128) void node_gemm_kernel(
    const float* __restrict__ A, const float* __restrict__ B,
    const float* __restrict__ bias, float* __restrict__ out, int nTiles) {
  __shared__ float Bc[128 * S68];   // column-major [n][k]
  __shared__ float At[TM * S68];    // row-major   [m][k]
  __shared__ float bs[128];

  const int t = threadIdx.x;
  for (int i = t; i < 64 * 128; i += 128) {
    int k = i >> 7, n = i & 127;
    Bc[n * S68 + k] = B[i];
  }
  bs[t] = bias[t];
  __syncthreads();

  const int lane = t & 31;
  const int wv = t >> 5;
  const int am = lane & 15;
  const int kh = (lane >> 4) << 1;
  const int rhi = (lane >> 4) << 3;

  for (int tile = blockIdx.x; tile < nTiles; tile += gridDim.x) {
    const long rBase = (long)tile * TM;
    for (int i = t; i < TM * 64; i += 128) {
      int r = i >> 6, k = i & 63;
      At[r * S68 + k] = A[(rBase + r) * 64 + k];
    }
    __syncthreads();

    const int n0 = wv * 32, n1 = n0 + 16;
    v8f acc0 = {}, acc1 = {};
#pragma unroll
    for (int k0 = 0; k0 < 64; k0 += 4) {
      v2f a  = *(const v2f*)&At[am * S68 + k0 + kh];
      v2f b0 = *(const v2f*)&Bc[(n0 + am) * S68 + k0 + kh];
      v2f b1 = *(const v2f*)&Bc[(n1 + am) * S68 + k0 + kh];
      acc0 = wmma4(a, b0, acc0);
      acc1 = wmma4(a, b1, acc1);
    }
#pragma unroll
    for (int v = 0; v < 8; ++v) {
      int r = v + rhi;
      int c0 = n0 + am, c1 = n1 + am;
      out[(rBase + r) * EH + c0] = acc0[v] + bs[c0];
      out[(rBase + r) * EH + c1] = acc1[v] + bs[c1];
    }
    __syncthreads();
  }
}

// ------------------------------------------- per-step edge kernel (in-place h)
// h[e] = relu( nodeW1[from[e]] + ef[e]@W2 + (s[from[e]] - h[e^1])@W3 + b2 + b3 )
__global__ __launch_bounds__(128) void edge_step_kernel(
    float* __restrict__ h, const float* __restrict__ s,
    const int* __restrict__ from, const float* __restrict__ nodeW1,
    const float* __restrict__ ef, const float* __restrict__ W2,
    const float* __restrict__ W3, const float* __restrict__ b2,
    const float* __restrict__ b3, int nTiles) {
  __shared__ float W3c[128 * S132];  // column-major [n][k], k<128
  __shared__ float W2c[128 * S20];   // column-major [n][k], k<16
  __shared__ float Aw[TM * S132];    // wu tile [m][k]
  __shared__ float Ae[TM * S20];     // edge-feature tile [m][k]
  __shared__ float nw[TM * S130];    // gathered nodeW1 tile
  __shared__ float bs[128];
  __shared__ int   sF[TM];

  const int t = threadIdx.x;
  for (int i = t; i < 128 * 128; i += 128) {
    int k = i >> 7, n = i & 127;
    W3c[n * S132 + k] = W3[i];
  }
  for (int i = t; i < 16 * 128; i += 128) {
    int k = i >> 7, n = i & 127;
    W2c[n * S20 + k] = W2[i];
  }
  bs[t] = b2[t] + b3[t];
  __syncthreads();

  const int lane = t & 31;
  const int wv = t >> 5;
  const int am = lane & 15;
  const int kh = (lane >> 4) << 1;
  const int rhi = (lane >> 4) << 3;

  for (int tile = blockIdx.x; tile < nTiles; tile += gridDim.x) {
    const long eBase = (long)tile * TM;
    if (t < TM) sF[t] = from[eBase + t];
    __syncthreads();

    // stage tile: wu = s[from] - h[backlink], nodeW1 gather, edge features
    {
      const int c = t;  // column 0..127
      for (int r = 0; r < TM; ++r) {
        float hv = h[(eBase + (r ^ 1)) * EH + c];       // backlink inside tile
        float sv = s[(long)sF[r] * EH + c];
        Aw[r * S132 + c] = sv - hv;
        nw[r * S130 + c] = nodeW1[(long)sF[r] * EH + c];
      }
      for (int i = t; i < TM * 16; i += 128) {
        int r = i >> 4, k = i & 15;
        Ae[r * S20 + k] = ef[(eBase + r) * 16 + k];
      }
    }
    __syncthreads();

    const int n0 = wv * 32, n1 = n0 + 16;
    v8f acc0 = {}, acc1 = {};
#pragma unroll
    for (int k0 = 0; k0 < 128; k0 += 4) {       // wu @ W3
      v2f a  = *(const v2f*)&Aw[am * S132 + k0 + kh];
      v2f b0 = *(const v2f*)&W3c[(n0 + am) * S132 + k0 + kh];
      v2f b1 = *(const v2f*)&W3c[(n1 + am) * S132 + k0 + kh];
      acc0 = wmma4(a, b0, acc0);
      acc1 = wmma4(a, b1, acc1);
    }
#pragma unroll
    for (int k0 = 0; k0 < 16; k0 += 4) {        // ef @ W2
      v2f a  = *(const v2f*)&Ae[am * S20 + k0 + kh];
      v2f b0 = *(const v2f*)&W2c[(n0 + am) * S20 + k0 + kh];
      v2f b1 = *(const v2f*)&W2c[(n1 + am) * S20 + k0 + kh];
      acc0 = wmma4(a, b0, acc0);
      acc1 = wmma4(a, b1, acc1);
    }

#pragma unroll
    for (int v = 0; v < 8; ++v) {
      int r = v + rhi;
      int c0 = n0 + am, c1 = n1 + am;
      float x0 = acc0[v] + nw[r * S130 + c0] + bs[c0];
      float x1 = acc1[v] + nw[r * S130 + c1] + bs[c1];
      h[(eBase + r) * EH + c0] = x0 > 0.0f ? x0 : 0.0f;
      h[(eBase + r) * EH + c1] = x1 > 0.0f ? x1 : 0.0f;
    }
    __syncthreads();
  }
}

// ------------------------------------------- final edge GEMM + scatter to nodes
// acc[to[e]] += h[e]@U2 + u2b
__global__ __launch_bounds__(128) void edge_final_kernel(
    const float* __restrict__ h, const int* __restrict__ to,
    const float* __restrict__ U2, const float* __restrict__ u2b,
    float* __restrict__ acc, int nTiles) {
  __shared__ float Bc[128 * S132];
  __shared__ float At[TM * S132];
  __shared__ float bs[128];
  __shared__ int   sT[TM];

  const int t = threadIdx.x;
  for (int i = t; i < 128 * 128; i += 128) {
    int k = i >> 7, n = i & 127;
    Bc[n * S132 + k] = U2[i];
  }
  bs[t] = u2b[t];
  __syncthreads();

  const int lane = t & 31;
  const int wv = t >> 5;
  const int am = lane & 15;
  const int kh = (lane >> 4) << 1;
  const int rhi = (lane >> 4) << 3;

  for (int tile = blockIdx.x; tile < nTiles; tile += gridDim.x) {
    const long eBase = (long)tile * TM;
    if (t < TM) sT[t] = to[eBase + t];
    __syncthreads();
    {
      const int c = t;
      for (int r = 0; r < TM; ++r) At[r * S132 + c] = h[(eBase + r) * EH + c];
    }
    __syncthreads();

    const int n0 = wv * 32, n1 = n0 + 16;
    v8f acc0 = {}, acc1 = {};
#pragma unroll
    for (int k0 = 0; k0 < 128; k0 += 4) {
      v2f a  = *(const v2f*)&At[am * S132 + k0 + kh];
      v2f b0 = *(const v2f*)&Bc[(n0 + am) * S132 + k0 + kh];
      v2f b1 = *(const v2f*)&Bc[(n1 + am) * S132 + k0 + kh];
      acc0 = wmma4(a, b0, acc0);
      acc1 = wmma4(a, b1, acc1);
    }
#pragma unroll
    for (int v = 0; v < 8; ++v) {
      int r = v + rhi;
      int c0 = n0 + am, c1 = n1 + am;
      atomicAdd(&acc[(long)sT[r] * EH + c0], acc0[v] + bs[c0]);
      atomicAdd(&acc[(long)sT[r] * EH + c1], acc1[v] + bs[c1]);
    }
    __syncthreads();
  }
}

// ---------------------------------------------------------------- launcher
extern "C" void kernel_launch(void* const* d_in, const int* in_sizes, int n_in,
                              void* d_out, int out_size, void* d_ws, size_t ws_size,
                              hipStream_t stream) {
  const float* node_features = (const float*)d_in[0];   // [N,64]
  const float* edge_features = (const float*)d_in[1];   // [E,16]
  const float* edge_hiddens  = (const float*)d_in[2];   // [E,128] (zeros)
  const int*   edges         = (const int*)d_in[3];     // [2,E]
  const float* W1_w = (const float*)d_in[4];
  const float* W1_b = (const float*)d_in[5];
  const float* W2_w = (const float*)d_in[6];
  const float* W2_b = (const float*)d_in[7];
  const float* W3_w = (const float*)d_in[8];
  const float* W3_b = (const float*)d_in[9];
  const float* U1_w = (const float*)d_in[10];
  const float* U1_b = (const float*)d_in[11];
  const float* U2_w = (const float*)d_in[12];
  const float* U2_b = (const float*)d_in[13];

  const int N = in_sizes[0] / 64;       // 100000
  const int E = in_sizes[1] / 16;       // 600000
  const int* from_nodes = edges;
  const int* to_nodes   = edges + E;

  float* h      = (float*)d_ws;                     // [E,128]
  float* sbuf   = h + (size_t)E * EH;               // [N,128] (also final acc)
  float* nodeW1 = sbuf + (size_t)N * EH;            // [N,128]

  const int nTilesN = N / TM;
  const int nTilesE = E / TM;
  const int gN = nTilesN < 1024 ? nTilesN : 1024;
  const int gE = nTilesE < 1024 ? nTilesE : 1024;
  const long nodeElems = (long)N * EH;
  const long edgeElems = (long)E * EH;

  // nodeW1 = nf @ W1 + b1
  node_gemm_kernel<<<gN, 128, 0, stream>>>(node_features, W1_w, W1_b, nodeW1, nTilesN);
  // h = edge_hiddens (inputs must not be mutated; h is updated in place)
  hipMemcpyAsync(h, edge_hiddens, (size_t)edgeElems * sizeof(float),
                 hipMemcpyDeviceToDevice, stream);

  for (int step = 0; step < 6; ++step) {
    zero_kernel<<<4096, 256, 0, stream>>>(sbuf, nodeElems);
    scatter_kernel<<<4096, 256, 0, stream>>>(h, from_nodes, sbuf, edgeElems);
    edge_step_kernel<<<gE, 128, 0, stream>>>(h, sbuf, from_nodes, nodeW1,
                                             edge_features, W2_w, W3_w,
                                             W2_b, W3_b, nTilesE);
  }

  // acc = nf @ U1 + b1  (reuse sbuf)
  node_gemm_kernel<<<gN, 128, 0, stream>>>(node_features, U1_w, U1_b, sbuf, nTilesN);
  // acc[to[e]] += h[e] @ U2 + b
  edge_final_kernel<<<gE, 128, 0, stream>>>(h, to_nodes, U2_w, U2_b, sbuf, nTilesE);
  // out = relu(acc)
  relu_kernel<<<4096, 256, 0, stream>>>(sbuf, (float*)d_out, nodeElems);
}